// LearningTanhModule_26379689132663
// MI455X (gfx1250) — compile-verified
//
#include <hip/hip_runtime.h>

typedef __attribute__((ext_vector_type(2))) float v2f;
typedef __attribute__((ext_vector_type(4))) float v4f;
typedef __attribute__((ext_vector_type(8))) float v8f;

// tanh via hardware exp2 + rcp: tanh(x) = (e-1)/(e+1), e = 2^(2*log2(e)*x).
// Clamp to +-9 so e stays finite; at |x|=9 the result already rounds to +-1.
__device__ __forceinline__ float fast_tanh(float x) {
    float y = fminf(fmaxf(x, -9.0f), 9.0f);
    float e = __builtin_amdgcn_exp2f(y * 2.885390081777927f); // 2*log2(e)
    return (e - 1.0f) * __builtin_amdgcn_rcpf(e + 1.0f);
}

__global__ __launch_bounds__(256) void tanh_amp_dot_kernel(
    const float* __restrict__ x,      // [n] inputs
    const float* __restrict__ w,      // [256] interleaved amp/bias
    float* __restrict__ out,          // [n] outputs
    int n)
{
    const int lane = threadIdx.x & 31;
    const int wave = blockIdx.x * (blockDim.x >> 5) + (threadIdx.x >> 5);
    const int base = wave * 16;                 // 16 rows per wave tile
    if (base >= n) return;                      // wave-uniform: EXEC stays full

    const int m = lane & 15;                    // row within tile / N column
    const int h = lane >> 4;                    // lane-half selects K pair (2h, 2h+1)

    if (base + 16 <= n) {
        // ---- full tile: f32 WMMA path ----
        const float xr = x[base + m];           // lanes 0-15 and 16-31 share rows
        v8f acc = {0.f, 0.f, 0.f, 0.f, 0.f, 0.f, 0.f, 0.f};

        #pragma unroll
        for (int c = 0; c < 32; ++c) {
            // per-lane j0 = 4c + 2h; contiguous 16B load gives
            // [amp[j0], bias[j0], amp[j0+1], bias[j0+1]]
            const v4f wv = *(const v4f*)(w + 8 * c + 4 * h);

            v2f a, b;
            // A (16x4 f32): VGPR0 -> K=2h, VGPR1 -> K=2h+1 (per ISA layout)
            a.x = fast_tanh(xr + wv.y);
            a.y = fast_tanh(xr + wv.w);
            // B (4x16 f32): same (lane,VGPR)->K mapping; broadcast over N
            b.x = wv.x;
            b.y = wv.z;

            // D = A x B + C, exact fp32 (V_WMMA_F32_16X16X4_F32)
            acc = __builtin_amdgcn_wmma_f32_16x16x4_f32(
                false, a, false, b, (short)0, acc, false, false);
        }

        // D layout: VGPR v holds M = v + 8h at N = m; all N columns identical.
        // Lane 0 writes rows 0-7, lane 16 writes rows 8-15.
        if (m == 0) {
            float* p = out + base + h * 8;
            v4f lo = {acc[0], acc[1], acc[2], acc[3]};
            v4f hi = {acc[4], acc[5], acc[6], acc[7]};
            *(v4f*)(p)     = lo;
            *(v4f*)(p + 4) = hi;
        }
    } else {
        // ---- ragged tail (unused for n % 16 == 0, kept for safety) ----
        const int row = base + lane;
        if (row < n) {
            const float xr = x[row];
            float s = 0.f;
            for (int j = 0; j < 128; ++j)
                s = fmaf(w[2 * j], fast_tanh(xr + w[2 * j + 1]), s);
            out[row] = s;
        }
    }
}

extern "C" void kernel_launch(void* const* d_in, const int* in_sizes, int n_in,
                              void* d_out, int out_size, void* d_ws, size_t ws_size,
                              hipStream_t stream) {
    const float* x = (const float*)d_in[0];   // input  (N,1) f32
    const float* w = (const float*)d_in[1];   // weight (256,1) f32 interleaved
    float* out = (float*)d_out;               // output (N,1) f32
    const int n = in_sizes[0];

    const int tiles  = (n + 15) / 16;         // one 16-row tile per wave
    const int blocks = (tiles + 7) / 8;       // 8 waves per 256-thread block
    tanh_amp_dot_kernel<<<blocks, 256, 0, stream>>>(x, w, out, n);
}